// FewShotYOLOLoss_60473139528404
// MI455X (gfx1250) — compile-verified
//
#include <hip/hip_runtime.h>

// ---------------- problem constants (from reference) ----------------
#define B_    16
#define M_    20
#define NC_   80
#define RM_   16
#define NO_   144
#define HW0_  6400
#define HW1_  1600
#define HW2_  400
#define A_    8400          // 6400+1600+400
#define PI_F  3.14159265358979323846f

typedef __attribute__((ext_vector_type(2))) float v2f;
typedef __attribute__((ext_vector_type(8))) float v8f;

// ---------------- workspace layout (float offsets) ----------------
#define OFF_BBOX   0L           // B*A*4      = 537600
#define OFF_S0     537600L      // B*A        = 134400   sigmoid(score class0)
#define OFF_SC0    672000L      // B*A        = 134400   raw logit class0
#define OFF_OV     806400L      // B*M*A      = 2688000  overlaps
#define OFF_AL     3494400L     // B*M*A      = 2688000  align
#define OFF_MP     6182400L     // B*M*A      = 2688000  mask_pos
#define OFF_PAL    8870400L     // B*M = 320  pos_align
#define OFF_POV    8870720L     // B*M = 320  pos_overlaps
#define OFF_FGM    8871040L     // B*A        fg mask (0/1 float)
#define OFF_TGT    9005440L     // B*A        target gt index (int)
#define OFF_PSP    9139840L     // 5250 partials: softplus sum
#define OFF_PW     9145090L     // 525 partials: sum ts (weights)
#define OFF_PCORR  9145615L     // 525 partials: sum s0*ts
#define OFF_PIOU   9146140L     // 525 partials: iou loss
#define OFF_PDFL   9146665L     // 525 partials: dfl loss

// ---------------- helpers ----------------
__device__ __forceinline__ void anchor_xy(int a, float& ax, float& ay) {
  int w, la;
  if (a < HW0_)            { w = 80; la = a; }
  else if (a < HW0_ + HW1_){ w = 40; la = a - HW0_; }
  else                     { w = 20; la = a - (HW0_ + HW1_); }
  ax = (float)(la % w) + 0.5f;
  ay = (float)(la / w) + 0.5f;
}

__device__ __forceinline__ const float* level_ptr(const float* f0, const float* f1,
                                                  const float* f2, int a, int& hw, int& la) {
  if (a < HW0_)        { hw = HW0_; la = a;               return f0; }
  if (a < HW0_ + HW1_) { hw = HW1_; la = a - HW0_;        return f1; }
  hw = HW2_; la = a - (HW0_ + HW1_);                      return f2;
}

// ciou(b1, b2) exactly as reference (eps placement preserved)
__device__ __forceinline__ float ciou_f(float b1x1, float b1y1, float b1x2, float b1y2,
                                        float b2x1, float b2y1, float b2x2, float b2y2) {
  const float eps = 1e-7f;
  float w1 = b1x2 - b1x1, h1 = b1y2 - b1y1 + eps;
  float w2 = b2x2 - b2x1, h2 = b2y2 - b2y1 + eps;
  float iw = fmaxf(fminf(b1x2, b2x2) - fmaxf(b1x1, b2x1), 0.f);
  float ih = fmaxf(fminf(b1y2, b2y2) - fmaxf(b1y1, b2y1), 0.f);
  float inter = iw * ih;
  float uni = w1 * h1 + w2 * h2 - inter + eps;
  float iou = inter / uni;
  float cw = fmaxf(b1x2, b2x2) - fminf(b1x1, b2x1);
  float ch = fmaxf(b1y2, b2y2) - fminf(b1y1, b2y1);
  float c2 = cw * cw + ch * ch + eps;
  float dx = b2x1 + b2x2 - b1x1 - b1x2;
  float dy = b2y1 + b2y2 - b1y1 - b1y2;
  float rho2 = (dx * dx + dy * dy) * 0.25f;
  float d = atanf(w2 / h2) - atanf(w1 / h1);
  float v = (4.0f / (PI_F * PI_F)) * d * d;
  float alpha = v / (v - iou + (1.0f + eps));
  return iou - (rho2 / c2 + v * alpha);
}

// ---------------- K1: DFL softmax decode with WMMA f32 16x16x4 ----------------
// Block = 256 threads = 64 anchors x 4 sides. Each thread computes a 16-bin
// softmax into LDS; each wave then feeds 2 groups of 16 rows through 4 chained
// V_WMMA_F32_16X16X4_F32 with B[k][n] = k (broadcast across N), yielding exact
// f32 expectations sum_k p_k * k on the matrix pipe.
__global__ __launch_bounds__(256) void k_decode(const float* __restrict__ f0,
                                                const float* __restrict__ f1,
                                                const float* __restrict__ f2,
                                                float* __restrict__ bbox,
                                                float* __restrict__ s0sig,
                                                float* __restrict__ sc0raw) {
  __shared__ float P[256][17];   // padded: conflict-free fragment gathers
  __shared__ float Q[256];
  const int t    = threadIdx.x;
  const int g    = blockIdx.x * 64 + (t >> 2);   // flat (b*A + a)
  const int side = t & 3;
  const int b = g / A_, a = g % A_;

  int hw, la;
  const float* fp = level_ptr(f0, f1, f2, a, hw, la);
  const float* p0 = fp + ((long)b * NO_ + side * RM_) * hw + la;

  float l[RM_];
  float mx = -1e30f;
#pragma unroll
  for (int k = 0; k < RM_; ++k) { l[k] = p0[(long)k * hw]; mx = fmaxf(mx, l[k]); }
  float s = 0.f;
#pragma unroll
  for (int k = 0; k < RM_; ++k) { float e = expf(l[k] - mx); P[t][k] = e; s += e; }
  float inv = 1.0f / s;
#pragma unroll
  for (int k = 0; k < RM_; ++k) P[t][k] *= inv;

  if (side == 0) {
    float sc = fp[((long)b * NO_ + 64) * hw + la];   // class-0 logit
    sc0raw[g] = sc;
    s0sig[g]  = 1.0f / (1.0f + expf(-sc));
  }
  __syncthreads();

  const int lane = t & 31;
  const int wave = t >> 5;
#pragma unroll
  for (int grp = 0; grp < 2; ++grp) {
    const int R = wave * 32 + grp * 16;            // 16 rows of this WMMA
    v8f acc = {0.f, 0.f, 0.f, 0.f, 0.f, 0.f, 0.f, 0.f};
#pragma unroll
    for (int c = 0; c < 4; ++c) {                  // K chunks of 4
      const int kbase = 4 * c;
      const int row = R + (lane & 15);
      const int khi = (lane < 16) ? 0 : 2;         // A/B f32 16x4 layout
      v2f Am, Bm;
      Am.x = P[row][kbase + khi];
      Am.y = P[row][kbase + khi + 1];
      Bm.x = (float)(kbase + khi);                 // proj[k] broadcast over N
      Bm.y = (float)(kbase + khi + 1);
      acc = __builtin_amdgcn_wmma_f32_16x16x4_f32(false, Am, false, Bm,
                                                  (short)0, acc, false, false);
    }
    // every column of D equals the row expectation: lane0 holds rows 0..7,
    // lane16 holds rows 8..15 (C/D layout: VGPR j -> M=j | M=j+8)
    if (lane == 0) {
      Q[R + 0] = acc[0]; Q[R + 1] = acc[1]; Q[R + 2] = acc[2]; Q[R + 3] = acc[3];
      Q[R + 4] = acc[4]; Q[R + 5] = acc[5]; Q[R + 6] = acc[6]; Q[R + 7] = acc[7];
    } else if (lane == 16) {
      Q[R + 8]  = acc[0]; Q[R + 9]  = acc[1]; Q[R + 10] = acc[2]; Q[R + 11] = acc[3];
      Q[R + 12] = acc[4]; Q[R + 13] = acc[5]; Q[R + 14] = acc[6]; Q[R + 15] = acc[7];
    }
  }
  __syncthreads();

  float dist = Q[t];
  float ax, ay; anchor_xy(a, ax, ay);
  float outv;
  if (side == 0)      outv = ax - dist;
  else if (side == 1) outv = ay - dist;
  else if (side == 2) outv = ax + dist;
  else                outv = ay + dist;
  bbox[(long)g * 4 + side] = outv;
}

// ---------------- K2: pairwise overlaps + align  (B*M*A) ----------------
__global__ __launch_bounds__(256) void k_pair(const float* __restrict__ gt,
                                              const int* __restrict__ mgt,
                                              const float* __restrict__ bbox,
                                              const float* __restrict__ s0sig,
                                              float* __restrict__ ov,
                                              float* __restrict__ al) {
  const long idx = (long)blockIdx.x * 256 + threadIdx.x;   // b*M*A + m*A + a
  const int a  = (int)(idx % A_);
  const int bm = (int)(idx / A_);
  const int m  = bm % M_, b = bm / M_;

  const float* gp = gt + (long)bm * 4;
  float cx = gp[0], cy = gp[1], hw2 = gp[2] * 0.5f, hh2 = gp[3] * 0.5f;
  float x1 = cx - hw2, y1 = cy - hh2, x2 = cx + hw2, y2 = cy + hh2;
  float mg = (mgt[bm] != 0) ? 1.0f : 0.0f;

  float ax, ay; anchor_xy(a, ax, ay);
  float mn = fminf(fminf(ax - x1, ay - y1), fminf(x2 - ax, y2 - ay));
  float mf = (mn > 1e-9f) ? mg : 0.0f;

  const float* pb = bbox + ((long)b * A_ + a) * 4;
  __builtin_prefetch(pb, 0, 0);                       // gfx1250 global_prefetch path
  float ci = ciou_f(x1, y1, x2, y2, pb[0], pb[1], pb[2], pb[3]);
  float o  = fmaxf(ci, 0.0f) * mf;
  float o2 = o * o;
  float o6 = o2 * o2 * o2;
  float av = sqrtf(s0sig[(long)b * A_ + a]) * o6 * mf;   // alpha=0.5, beta=6
  ov[idx] = o;
  al[idx] = av;
}

// ---------------- K3: exact-tie top-10 per (b,m) row ----------------
__global__ __launch_bounds__(256) void k_topk(const float* __restrict__ al,
                                              const int* __restrict__ mgt,
                                              const float* __restrict__ gt,
                                              float* __restrict__ mp) {
  const int bm = blockIdx.x;
  if (mgt[bm] == 0) return;                      // masked gt -> row stays zero
  __shared__ float sv[2560];
  __shared__ int   si[2560];
  const int tid = threadIdx.x;
  const long base = (long)bm * A_;

  float tv[10]; int ti[10];
#pragma unroll
  for (int j = 0; j < 10; ++j) { tv[j] = -1.0f; ti[j] = 0x7fffffff; }
  for (int a = tid; a < A_; a += 256) {
    float v = al[base + a];
    if (v > tv[9] || (v == tv[9] && a < ti[9])) {
      tv[9] = v; ti[9] = a;
#pragma unroll
      for (int j = 9; j > 0; --j) {
        bool sw = (tv[j] > tv[j-1]) || (tv[j] == tv[j-1] && ti[j] < ti[j-1]);
        if (sw) { float fv = tv[j]; tv[j] = tv[j-1]; tv[j-1] = fv;
                  int  iv = ti[j]; ti[j] = ti[j-1]; ti[j-1] = iv; }
      }
    }
  }
#pragma unroll
  for (int j = 0; j < 10; ++j) { sv[tid * 10 + j] = tv[j]; si[tid * 10 + j] = ti[j]; }
  __syncthreads();

  if (tid == 0) {
    const float* gp = gt + (long)bm * 4;
    float hw2 = gp[2] * 0.5f, hh2 = gp[3] * 0.5f;
    float x1 = gp[0] - hw2, y1 = gp[1] - hh2, x2 = gp[0] + hw2, y2 = gp[1] + hh2;
    for (int r = 0; r < 10; ++r) {
      int bsl = 0; float bv = sv[0]; int bi = si[0];
      for (int j = 1; j < 2560; ++j) {
        float v = sv[j]; int i2 = si[j];
        if (v > bv || (v == bv && i2 < bi)) { bv = v; bi = i2; bsl = j; }
      }
      sv[bsl] = -2.0f; si[bsl] = 0x7fffffff;
      float ax, ay; anchor_xy(bi, ax, ay);
      float mn = fminf(fminf(ax - x1, ay - y1), fminf(x2 - ax, y2 - ay));
      if (mn > 1e-9f) mp[base + bi] = 1.0f;      // mask_topk * mask_full
    }
  }
}

// ---------------- K4: conflict resolution, fg mask, target index ----------------
__global__ __launch_bounds__(256) void k_resolve(const float* __restrict__ ov,
                                                 float* __restrict__ mp,
                                                 float* __restrict__ fgm,
                                                 int* __restrict__ tgt) {
  const int ba = blockIdx.x * 256 + threadIdx.x;   // b*A + a
  const int b = ba / A_, a = ba % A_;
  const long col = (long)b * M_ * A_ + a;

  float fg = 0.f; int first = -1;
#pragma unroll
  for (int m = 0; m < M_; ++m) {
    float v = mp[col + (long)m * A_];
    fg += v;
    if (v > 0.f && first < 0) first = m;
  }
  if (fg > 1.0f) {                                 // argmax_m overlaps (first max)
    float bo = ov[col]; int bmx = 0;
#pragma unroll
    for (int m = 1; m < M_; ++m) {
      float v = ov[col + (long)m * A_];
      if (v > bo) { bo = v; bmx = m; }
    }
#pragma unroll
    for (int m = 0; m < M_; ++m) mp[col + (long)m * A_] = (m == bmx) ? 1.0f : 0.0f;
    first = bmx; fg = 1.0f;
  }
  fgm[ba] = (fg > 0.f) ? 1.0f : 0.0f;
  tgt[ba] = (first >= 0) ? first : 0;
}

// ---------------- K5: pos_align / pos_overlaps per (b,m) ----------------
__global__ __launch_bounds__(256) void k_rowmax(const float* __restrict__ al,
                                                const float* __restrict__ ov,
                                                const float* __restrict__ mp,
                                                float* __restrict__ posal,
                                                float* __restrict__ posov) {
  const int bm = blockIdx.x;
  const long base = (long)bm * A_;
  const int tid = threadIdx.x;
  float ma = 0.f, mo = 0.f;
  for (int a = tid; a < A_; a += 256) {
    float p = mp[base + a];
    if (p > 0.f) {
      ma = fmaxf(ma, al[base + a] * p);
      mo = fmaxf(mo, ov[base + a] * p);
    }
  }
  __shared__ float ra[256], ro[256];
  ra[tid] = ma; ro[tid] = mo; __syncthreads();
  for (int s = 128; s > 0; s >>= 1) {
    if (tid < s) { ra[tid] = fmaxf(ra[tid], ra[tid + s]); ro[tid] = fmaxf(ro[tid], ro[tid + s]); }
    __syncthreads();
  }
  if (tid == 0) { posal[bm] = ra[0]; posov[bm] = ro[0]; }
}

// ---------------- K6: softplus(score) over all B*NC*A logits ----------------
__global__ __launch_bounds__(256) void k_cls(const float* __restrict__ f0,
                                             const float* __restrict__ f1,
                                             const float* __restrict__ f2,
                                             float* __restrict__ psp) {
  const int tid = threadIdx.x;
  const long base = (long)blockIdx.x * 2048;
  float acc = 0.f;
#pragma unroll
  for (int j = 0; j < 8; ++j) {
    long e = base + tid + (long)j * 256;
    const float* f; long r; int hw;
    if (e < 8192000L)        { f = f0; r = e;             hw = HW0_; }
    else if (e < 10240000L)  { f = f1; r = e - 8192000L;  hw = HW1_; }
    else                     { f = f2; r = e - 10240000L; hw = HW2_; }
    long per_b = (long)NC_ * hw;
    int b  = (int)(r / per_b);
    long r2 = r % per_b;
    int c   = (int)(r2 / hw);
    int pos = (int)(r2 % hw);
    float s = f[((long)b * NO_ + 64 + c) * hw + pos];
    acc += fmaxf(s, 0.f) + log1pf(expf(-fabsf(s)));      // softplus
  }
  __shared__ float red[256];
  red[tid] = acc; __syncthreads();
  for (int s = 128; s > 0; s >>= 1) { if (tid < s) red[tid] += red[tid + s]; __syncthreads(); }
  if (tid == 0) psp[blockIdx.x] = red[0];
}

// ---------------- K7: fused norm / weights / iou / dfl / cls-correction ----------------
__global__ __launch_bounds__(256) void k_main(const float* __restrict__ f0,
                                              const float* __restrict__ f1,
                                              const float* __restrict__ f2,
                                              const float* __restrict__ gt,
                                              const float* __restrict__ bbox,
                                              const float* __restrict__ sc0raw,
                                              const float* __restrict__ al,
                                              const float* __restrict__ mp,
                                              const float* __restrict__ posal,
                                              const float* __restrict__ posov,
                                              const float* __restrict__ fgm,
                                              const int* __restrict__ tgt,
                                              float* __restrict__ pw,
                                              float* __restrict__ pcorr,
                                              float* __restrict__ piou,
                                              float* __restrict__ pdfl) {
  const int ba = blockIdx.x * 256 + threadIdx.x;
  const int b = ba / A_, a = ba % A_;
  const long col = (long)b * M_ * A_ + a;

  // norm[b,a] = max_m align*mask_pos*pos_ov/(pos_align+eps)
  float nm = 0.f;
#pragma unroll
  for (int m = 0; m < M_; ++m) {
    float p = mp[col + (long)m * A_];
    if (p > 0.f) {
      int bm = b * M_ + m;
      float v = al[col + (long)m * A_] * p * posov[bm] / (posal[bm] + 1e-9f);
      nm = fmaxf(nm, v);
    }
  }
  const float fg = fgm[ba];
  const float w  = fg * nm;                    // ts.sum(-1)
  const float corr = sc0raw[ba] * w;           // sum_c s*ts  (only class 0)
  float li = 0.f, ld = 0.f;

  if (fg > 0.f && w > 0.f) {
    const int tm = tgt[ba];
    const float* gp = gt + ((long)b * M_ + tm) * 4;
    float hw2 = gp[2] * 0.5f, hh2 = gp[3] * 0.5f;
    float tx1 = gp[0] - hw2, ty1 = gp[1] - hh2, tx2 = gp[0] + hw2, ty2 = gp[1] + hh2;
    const float* pb = bbox + (long)ba * 4;
    float ci = ciou_f(pb[0], pb[1], pb[2], pb[3], tx1, ty1, tx2, ty2);
    li = (1.0f - ci) * w;

    float ax, ay; anchor_xy(a, ax, ay);
    float tv4[4] = { ax - tx1, ay - ty1, tx2 - ax, ty2 - ay };
    int hw, la;
    const float* fp = level_ptr(f0, f1, f2, a, hw, la);
    float dsum = 0.f;
#pragma unroll
    for (int side = 0; side < 4; ++side) {
      float tt = fminf(fmaxf(tv4[side], 0.f), (float)RM_ - 1.0f - 0.01f);
      int tl = (int)tt;
      float wl = (float)(tl + 1) - tt;
      float wr = 1.0f - wl;
      const float* p0 = fp + ((long)b * NO_ + side * RM_) * hw + la;
      float l[RM_];
      float mx = -1e30f;
#pragma unroll
      for (int k = 0; k < RM_; ++k) { l[k] = p0[(long)k * hw]; mx = fmaxf(mx, l[k]); }
      float ssum = 0.f;
#pragma unroll
      for (int k = 0; k < RM_; ++k) ssum += expf(l[k] - mx);
      float lse = logf(ssum) + mx;
      float lptl = l[tl]     - lse;
      float lptr = l[tl + 1] - lse;
      dsum += -(lptl * wl + lptr * wr);
    }
    ld = dsum * 0.25f * w;
  }

  __shared__ float r0[256], r1[256], r2[256], r3[256];
  const int tid = threadIdx.x;
  r0[tid] = w; r1[tid] = corr; r2[tid] = li; r3[tid] = ld;
  __syncthreads();
  for (int s = 128; s > 0; s >>= 1) {
    if (tid < s) {
      r0[tid] += r0[tid + s]; r1[tid] += r1[tid + s];
      r2[tid] += r2[tid + s]; r3[tid] += r3[tid + s];
    }
    __syncthreads();
  }
  if (tid == 0) {
    pw[blockIdx.x]    = r0[0];
    pcorr[blockIdx.x] = r1[0];
    piou[blockIdx.x]  = r2[0];
    pdfl[blockIdx.x]  = r3[0];
  }
}

// ---------------- K8: deterministic finalize ----------------
__global__ __launch_bounds__(256) void k_final(const float* __restrict__ psp,
                                               const float* __restrict__ pw,
                                               const float* __restrict__ pcorr,
                                               const float* __restrict__ piou,
                                               const float* __restrict__ pdfl,
                                               float* __restrict__ out) {
  const int tid = threadIdx.x;
  __shared__ float red[256];
  __shared__ float res[5];
  const float* arrs[5] = { psp, pw, pcorr, piou, pdfl };
  const int    cnts[5] = { 5250, 525, 525, 525, 525 };
  for (int p = 0; p < 5; ++p) {
    float acc = 0.f;
    for (int j = tid; j < cnts[p]; j += 256) acc += arrs[p][j];
    red[tid] = acc; __syncthreads();
    for (int s = 128; s > 0; s >>= 1) { if (tid < s) red[tid] += red[tid + s]; __syncthreads(); }
    if (tid == 0) res[p] = red[0];
    __syncthreads();
  }
  if (tid == 0) {
    float sp = res[0], wsum = res[1], corr = res[2], iou = res[3], dfl = res[4];
    float tss = fmaxf(wsum, 1.0f);
    out[0] = 7.5f * iou / tss + 0.5f * (sp - corr) / tss + 1.5f * dfl / tss;
  }
}

// ---------------- launcher ----------------
extern "C" void kernel_launch(void* const* d_in, const int* in_sizes, int n_in,
                              void* d_out, int out_size, void* d_ws, size_t ws_size,
                              hipStream_t stream) {
  const float* f0  = (const float*)d_in[0];
  const float* f1  = (const float*)d_in[1];
  const float* f2  = (const float*)d_in[2];
  const float* gt  = (const float*)d_in[3];
  const int*   mgt = (const int*)d_in[4];
  float* out = (float*)d_out;
  float* w   = (float*)d_ws;

  float* bbox   = w + OFF_BBOX;
  float* s0sig  = w + OFF_S0;
  float* sc0raw = w + OFF_SC0;
  float* ov     = w + OFF_OV;
  float* al     = w + OFF_AL;
  float* mp     = w + OFF_MP;
  float* posal  = w + OFF_PAL;
  float* posov  = w + OFF_POV;
  float* fgm    = w + OFF_FGM;
  int*   tgt    = (int*)(w + OFF_TGT);
  float* psp    = w + OFF_PSP;
  float* pw     = w + OFF_PW;
  float* pcorr  = w + OFF_PCORR;
  float* piou   = w + OFF_PIOU;
  float* pdfl   = w + OFF_PDFL;

  // mask_pos must start at zero every call (graph-capture-safe memset)
  hipMemsetAsync((void*)mp, 0, (size_t)B_ * M_ * A_ * sizeof(float), stream);

  k_decode <<<2100, 256, 0, stream>>>(f0, f1, f2, bbox, s0sig, sc0raw);
  k_pair   <<<10500, 256, 0, stream>>>(gt, mgt, bbox, s0sig, ov, al);
  k_topk   <<<B_ * M_, 256, 0, stream>>>(al, mgt, gt, mp);
  k_resolve<<<525, 256, 0, stream>>>(ov, mp, fgm, tgt);
  k_rowmax <<<B_ * M_, 256, 0, stream>>>(al, ov, mp, posal, posov);
  k_cls    <<<5250, 256, 0, stream>>>(f0, f1, f2, psp);
  k_main   <<<525, 256, 0, stream>>>(f0, f1, f2, gt, bbox, sc0raw, al, mp,
                                     posal, posov, fgm, tgt, pw, pcorr, piou, pdfl);
  k_final  <<<1, 256, 0, stream>>>(psp, pw, pcorr, piou, pdfl, out);
}